// DigitCapsuleLayer_43997644980829
// MI455X (gfx1250) — compile-verified
//
#include <hip/hip_runtime.h>
#include <math.h>

typedef float v2f __attribute__((ext_vector_type(2)));
typedef float v8f __attribute__((ext_vector_type(8)));

#define B_    256
#define R_    1152
#define C_    10
#define O_    16
#define I_    8
#define K_TOT (R_ * I_)       // 9216 : flattened (r,i) contraction dim for pass 1
#define N_    (C_ * O_)       // 160  : flattened (c,o) output dim
#define KSPLIT 18
#define KCHUNK (K_TOT / KSPLIT) // 512 (= 64 routes per K-chunk)
#define SVO   (B_ * N_)       // 40960 elements of s / v / out
#define BC    (R_ * C_)       // 11520 elements of b_ij / c_ij

// ---------------------------------------------------------------------------
// Zero the routing logits b_ij (must happen every launch: ws is not re-poisoned)
__global__ void zero_b_kernel(float* __restrict__ bij) {
    int t = blockIdx.x * blockDim.x + threadIdx.x;
    if (t < BC) bij[t] = 0.0f;
}

// ---------------------------------------------------------------------------
// c_ij = softmax over r (axis 0) of b_ij [R,C]. One block, wave w owns class w.
__global__ void softmax_kernel(const float* __restrict__ bij, float* __restrict__ cij) {
    const int tid  = threadIdx.x;
    const int c    = tid >> 5;      // wave index -> class
    const int lane = tid & 31;
    if (c >= C_) return;
    float mx = -3.402823466e38f;
    for (int r = lane; r < R_; r += 32) mx = fmaxf(mx, bij[r * C_ + c]);
    for (int m = 16; m >= 1; m >>= 1) mx = fmaxf(mx, __shfl_xor(mx, m, 32));
    float sum = 0.0f;
    for (int r = lane; r < R_; r += 32) sum += __expf(bij[r * C_ + c] - mx);
    for (int m = 16; m >= 1; m >>= 1) sum += __shfl_xor(sum, m, 32);
    const float inv = 1.0f / sum;
    for (int r = lane; r < R_; r += 32)
        cij[r * C_ + c] = __expf(bij[r * C_ + c] - mx) * inv;
}

// ---------------------------------------------------------------------------
// Pass 1: s[b,(c,o)] = sum_{(r,i)} u[b,(r,i)] * (c_ij[r,c] * W[r,c,o,i])
// One wave per 16x16 tile, K split 18 ways; partials to ws (deterministic).
// WMMA f32 16x16x4: A lane layout M=lane&15, K=k0+j+2*(lane>=16); B symmetric.
__global__ void __launch_bounds__(32)
gemm_s_kernel(const float* __restrict__ u, const float* __restrict__ W,
              const float* __restrict__ cij, float* __restrict__ part) {
    const int b0   = blockIdx.x << 4;   // batch tile (16 rows)
    const int c    = blockIdx.y;        // class == N-tile (N = c*16 + o)
    const int kz   = blockIdx.z;        // K-split slice
    const int lane = threadIdx.x;
    const int half = lane >> 4;
    const int mrow = lane & 15;

    v8f acc = {0.f,0.f,0.f,0.f,0.f,0.f,0.f,0.f};
    const float* uRow = u + (size_t)(b0 + mrow) * K_TOT;      // A row
    const float* Wco  = W + c * (O_ * I_) + mrow * I_;        // B col (o = mrow)
    const int kbase = kz * KCHUNK;
    for (int k0 = kbase; k0 < kbase + KCHUNK; k0 += 4) {
        const int kk = k0 + 2 * half;          // this lane's base K (j=0)
        const int r  = kk >> 3;                // same r for all 4 K's of the step
        const int i  = kk & 7;                 // even -> float2 aligned
        v2f a2 = *(const v2f*)(uRow + kk);                         // u[b, K..K+1]
        v2f w2 = *(const v2f*)(Wco + (size_t)r * (C_ * O_ * I_) + i); // W[r,c,o,i..i+1]
        const float cr = cij[r * C_ + c];      // routing coeff (scalar per step)
        v2f b2; b2.x = w2.x * cr; b2.y = w2.y * cr;
        acc = __builtin_amdgcn_wmma_f32_16x16x4_f32(
            false, a2, false, b2, (short)0, acc, false, false);
    }
    // D layout: VGPR v, lanes0-15 -> M=v, lanes16-31 -> M=v+8 ; col = mrow
    float* prow = part + (size_t)kz * SVO + (size_t)(b0 + 8 * half) * N_ + c * O_ + mrow;
#pragma unroll
    for (int vv = 0; vv < 8; ++vv) prow[(size_t)vv * N_] = acc[vv];
}

// ---------------------------------------------------------------------------
// Reduce K-split partials (fixed order -> deterministic) + bug-faithful squash:
// v = s^2*s / ((1+s^2)*sqrt(s^2)) == s*|s|/(1+s^2)
__global__ void reduce_squash_kernel(const float* __restrict__ part,
                                     float* __restrict__ v,
                                     float* __restrict__ out, int write_out) {
    const int t = blockIdx.x * blockDim.x + threadIdx.x;
    if (t >= SVO) return;
    float x = 0.0f;
#pragma unroll
    for (int kz = 0; kz < KSPLIT; ++kz) x += part[(size_t)kz * SVO + t];
    const float val = x * fabsf(x) / (1.0f + x * x);
    v[t] = val;
    if (write_out) out[t] = val;
}

// ---------------------------------------------------------------------------
// Pass 2: b[r,c] += (1/B) * sum_{i,o} W[r,c,o,i] * (sum_b u[b,r,i] * v[b,c,o]).
// Per block: r-pair (rows 0-7 -> r0 i=0..7, rows 8-15 -> r0+1) x one class.
// P = u_r^T (16xB) @ v (Bx16) via WMMA K=256, then W-dot + 16-lane reduce.
__global__ void __launch_bounds__(32)
pass2_kernel(const float* __restrict__ u, const float* __restrict__ W,
             const float* __restrict__ vbuf, float* __restrict__ bij) {
    const int r0   = blockIdx.x << 1;
    const int c    = blockIdx.y;
    const int lane = threadIdx.x;
    const int half = lane >> 4;
    const int mrow = lane & 15;

    v8f acc = {0.f,0.f,0.f,0.f,0.f,0.f,0.f,0.f};
    const float* uBase = u + r0 * I_ + mrow;      // A[m=mrow][K=b] = u[b, r0 + m/8, m%8]
    const float* vBase = vbuf + c * O_ + mrow;    // B[K=b][n=o=mrow] = v[b,c,o]
    for (int k0 = 0; k0 < B_; k0 += 4) {
        const int kk = k0 + 2 * half;
        v2f a2, b2;
        a2.x = uBase[(size_t)kk * K_TOT];
        a2.y = uBase[(size_t)(kk + 1) * K_TOT];
        b2.x = vBase[(size_t)kk * N_];
        b2.y = vBase[(size_t)(kk + 1) * N_];
        acc = __builtin_amdgcn_wmma_f32_16x16x4_f32(
            false, a2, false, b2, (short)0, acc, false, false);
    }
    // acc[vv] = P[m = vv + 8*half][o = mrow]  (i = vv, r = r0 + half)
    const float* wp = W + (size_t)(r0 + half) * (C_ * O_ * I_) + c * (O_ * I_) + mrow * I_;
    float partial = 0.0f;
#pragma unroll
    for (int vv = 0; vv < 8; ++vv) partial += wp[vv] * acc[vv];
    // reduce over o (= mrow) within each 16-lane half
    for (int m = 8; m >= 1; m >>= 1) partial += __shfl_xor(partial, m, 16);
    if (mrow == 0)  // single writer per (r,c): deterministic plain RMW
        bij[(r0 + half) * C_ + c] += partial * (1.0f / 256.0f);
}

// ---------------------------------------------------------------------------
extern "C" void kernel_launch(void* const* d_in, const int* in_sizes, int n_in,
                              void* d_out, int out_size, void* d_ws, size_t ws_size,
                              hipStream_t stream) {
    const float* u = (const float*)d_in[0];   // [B, R, I]
    const float* W = (const float*)d_in[1];   // [1, R, C, O, I]
    float* out = (float*)d_out;               // [B, C, O, 1] flat
    float* ws  = (float*)d_ws;

    float* part = ws;                          // KSPLIT * SVO = 737280 floats
    float* v    = part + (size_t)KSPLIT * SVO; // 40960
    float* cij  = v + SVO;                     // 11520
    float* bij  = cij + BC;                    // 11520  (~3.2 MB total)

    zero_b_kernel<<<(BC + 255) / 256, 256, 0, stream>>>(bij);
    for (int it = 0; it < 3; ++it) {
        softmax_kernel<<<1, 320, 0, stream>>>(bij, cij);
        gemm_s_kernel<<<dim3(B_ / 16, C_, KSPLIT), 32, 0, stream>>>(u, W, cij, part);
        reduce_squash_kernel<<<(SVO + 255) / 256, 256, 0, stream>>>(part, v, out, it == 2);
        if (it < 2)
            pass2_kernel<<<dim3(R_ / 2, C_), 32, 0, stream>>>(u, W, v, bij);
    }
}